// m3_33002528702905
// MI455X (gfx1250) — compile-verified
//
#include <hip/hip_runtime.h>

// Problem constants from the reference: x[B,T], a/b/c/w[H], e[1] -> z[B]
#define Bsz  64
#define Tlen 4096
#define Hdim 512

typedef float v2f __attribute__((ext_vector_type(2)));
typedef float v8f __attribute__((ext_vector_type(8)));

// ---------------------------------------------------------------------------
// Kernel 1: closed-form precompute.
//   Blocks 0..15 (256 thr): K[t] = sum_j (w_j*b_j) * a_j^(T-1-t)
//                           computed as exp2((T-1-t) * log2(a_j))  (v_exp/v_log)
//   Block 16: C = sum_j w_j*c_j*(1-a_j^T)/(1-a_j) + e
// ---------------------------------------------------------------------------
__global__ __launch_bounds__(256) void k1_precompute(
    const float* __restrict__ a, const float* __restrict__ b,
    const float* __restrict__ c, const float* __restrict__ w,
    const float* __restrict__ e, float* __restrict__ Kout,
    float* __restrict__ Cout)
{
  const int tid = threadIdx.x;
  if (blockIdx.x < 16) {
    __shared__ float wb[Hdim];
    __shared__ float lg[Hdim];
    for (int j = tid; j < Hdim; j += 256) {
      wb[j] = w[j] * b[j];
      lg[j] = __builtin_amdgcn_logf(a[j]);   // v_log_f32 = log2
    }
    __syncthreads();
    const int   t = blockIdx.x * 256 + tid;  // 16*256 = 4096 = Tlen
    const float p = (float)(Tlen - 1 - t);
    float acc = 0.f;
#pragma unroll 4
    for (int j = 0; j < Hdim; ++j)
      acc += wb[j] * __builtin_amdgcn_exp2f(p * lg[j]);  // v_exp_f32 = 2^x
    Kout[t] = acc;
  } else {
    __shared__ float red[256];
    float s = 0.f;
    for (int j = tid; j < Hdim; j += 256) {
      const float aj  = a[j];
      const float aT  = __builtin_amdgcn_exp2f(4096.f * __builtin_amdgcn_logf(aj));
      const float geo = (1.f - aT) / (1.f - aj);   // a in [0.95,0.999): safe
      s += w[j] * c[j] * geo;
    }
    red[tid] = s;
    __syncthreads();
    for (int off = 128; off > 0; off >>= 1) {
      if (tid < off) red[tid] += red[tid + off];
      __syncthreads();
    }
    if (tid == 0) Cout[0] = red[0] + e[0];
  }
}

// ---------------------------------------------------------------------------
// Kernel 2: split-K GEMV  z_partial = X @ K  via V_WMMA_F32_16X16X4_F32.
//   64 blocks x 128 threads (4 waves, wave32). Block owns K-chunk of 64.
//   Wave w owns rows [16w, 16w+16).  ISA A layout (32-bit A 16x4):
//     lanes 0-15 : M=lane,    VGPR0/1 = K=0,1
//     lanes 16-31: M=lane-16, VGPR0/1 = K=2,3
//   B rows (K index) are filled with K[k] replicated across all N columns, so
//   every column of D equals the partial GEMV result.  D layout: lane 0 holds
//   rows 0..7 (VGPR 0..7) of column 0, lane 16 holds rows 8..15.
// ---------------------------------------------------------------------------
__global__ __launch_bounds__(128) void k2_gemv_wmma(
    const float* __restrict__ X, const float* __restrict__ Kg,
    float* __restrict__ partial)
{
  __shared__ __align__(16) float kk[64];
  const int tid = threadIdx.x;
  if (tid < 64) kk[tid] = Kg[blockIdx.x * 64 + tid];
  __syncthreads();

  const int lane = tid & 31;          // wave32
  const int wv   = tid >> 5;          // 4 waves
  const int half = lane >> 4;         // 0: K=0,1   1: K=2,3
  const int lrow = lane & 15;         // M within 16-row tile
  const int row  = wv * 16 + lrow;

  const float* xr = X + (size_t)row * Tlen + (size_t)blockIdx.x * 64 + half * 2;

  v8f acc = {};
#pragma unroll
  for (int it = 0; it < 16; ++it) {
    v2f av = *(const v2f*)(xr + it * 4);                 // global_load_b64
    v2f bv = *(const v2f*)(&kk[it * 4 + half * 2]);      // ds_load_b64
    acc = __builtin_amdgcn_wmma_f32_16x16x4_f32(
        /*neg_a=*/false, av, /*neg_b=*/false, bv,
        /*c_mod=*/(short)0, acc, /*reuse_a=*/false, /*reuse_b=*/false);
  }

  // Column 0 of D: lane 0 -> rows 0..7, lane 16 -> rows 8..15 (per wave tile).
  if (lrow == 0) {
    const int base = blockIdx.x * 64 + wv * 16 + half * 8;
#pragma unroll
    for (int r = 0; r < 8; ++r) partial[base + r] = acc[r];
  }
}

// ---------------------------------------------------------------------------
// Kernel 3: deterministic final reduction over the 64 K-chunks, add C.
// ---------------------------------------------------------------------------
__global__ __launch_bounds__(64) void k3_reduce(
    const float* __restrict__ partial, const float* __restrict__ Cp,
    float* __restrict__ z)
{
  const int b = threadIdx.x;   // 64 outputs
  float s = Cp[0];
#pragma unroll 8
  for (int p = 0; p < 64; ++p) s += partial[p * 64 + b];
  z[b] = s;
}

// ---------------------------------------------------------------------------
extern "C" void kernel_launch(void* const* d_in, const int* in_sizes, int n_in,
                              void* d_out, int out_size, void* d_ws, size_t ws_size,
                              hipStream_t stream) {
  (void)in_sizes; (void)n_in; (void)out_size; (void)ws_size;
  const float* x = (const float*)d_in[0];   // [64, 4096]
  const float* a = (const float*)d_in[1];   // [512]
  const float* b = (const float*)d_in[2];   // [512]
  const float* c = (const float*)d_in[3];   // [512]
  const float* w = (const float*)d_in[4];   // [512]
  const float* e = (const float*)d_in[5];   // [1]
  float* z = (float*)d_out;                 // [64]

  float* wsf     = (float*)d_ws;
  float* Kg      = wsf;                     // 4096 floats
  float* Cp      = wsf + 4096;              // 1 float
  float* partial = wsf + 4160;              // 64 blocks * 64 rows

  k1_precompute<<<dim3(17), dim3(256), 0, stream>>>(a, b, c, w, e, Kg, Cp);
  k2_gemv_wmma <<<dim3(64), dim3(128), 0, stream>>>(x, Kg, partial);
  k3_reduce    <<<dim3(1),  dim3(64),  0, stream>>>(partial, Cp, z);
}